// Attention_42039139894154
// MI455X (gfx1250) — compile-verified
//
#include <hip/hip_runtime.h>

// ---------------------------------------------------------------------------
// Non-local attention block (no softmax) for MI455X / gfx1250, fp32 WMMA.
//
// Reassociation: y = (1/N) (G P^T) T  and  W_w y = (W_w S~) T, so the
// [B,4096,4096] attention map is never formed. W_b cancels under BN.
// All big GEMMs use V_WMMA_F32_16X16X4_F32 (exact fp32).
// ---------------------------------------------------------------------------

typedef __attribute__((ext_vector_type(2))) float v2f;
typedef __attribute__((ext_vector_type(8))) float v8f;

#define DEVI __device__ __forceinline__

DEVI v8f wmma4(v2f a, v2f b, v8f c) {
    // D = A(16x4) * B(4x16) + C(16x16), fp32 exact
    return __builtin_amdgcn_wmma_f32_16x16x4_f32(
        false, a, false, b, (short)0, c, false, false);
}

static constexpr int BB = 8, CI = 256, CO = 128, NN = 4096;

// workspace layout (float offsets)
static constexpr size_t OFF_G     = 0;                       // [8][128][4096]
static constexpr size_t OFF_T     = OFF_G + (size_t)BB*CO*NN;
static constexpr size_t OFF_P     = OFF_T + (size_t)BB*CO*NN;
static constexpr size_t OFF_SPART = OFF_P + (size_t)BB*CO*NN;      // [8][8][128][128]
static constexpr size_t OFF_S     = OFF_SPART + (size_t)BB*8*CO*CO; // [8][128][128]
static constexpr size_t OFF_M2    = OFF_S + (size_t)BB*CO*CO;      // [8][256][128]
static constexpr size_t OFF_U     = OFF_M2 + (size_t)BB*CI*CO;     // [8][256][4096]
static constexpr size_t OFF_BNP   = OFF_U + (size_t)BB*CI*NN;      // [512][256][2]
static constexpr size_t OFF_COEF  = OFF_BNP + (size_t)512*CI*2;    // [256][2]

// ---------------------------------------------------------------------------
// K1: fused projections  G/T/P[b] = w @ x[b] + bias    (M=128,K=256,N=4096)
// X tile [256 x 64] staged in LDS, pair-interleaved over channel c so that
// a B-fragment (rows c+2h, c+2h+1 at one column) is a single ds_load_b64.
// LDS idx(c,j) = (c>>1)*128 + j*2 + (c&1); size = 16384 floats = 64 KB.
// ---------------------------------------------------------------------------
__global__ __launch_bounds__(256) void proj_kernel(
    const float* __restrict__ x,
    const float* __restrict__ gw, const float* __restrict__ gbias,
    const float* __restrict__ tw, const float* __restrict__ tbias,
    const float* __restrict__ pw, const float* __restrict__ pbias,
    float* __restrict__ G, float* __restrict__ T, float* __restrict__ P)
{
    extern __shared__ float lx[];
    const int b = blockIdx.y, n0 = blockIdx.x * 64, tid = threadIdx.x;
    const float* xb = x + (size_t)b * CI * NN + n0;

    #pragma unroll
    for (int i = 0; i < 16; ++i) {                 // 4096 float4 loads total
        int f4 = tid + 256 * i;
        int c  = f4 >> 4;                          // channel row 0..255
        int j4 = f4 & 15;                          // col/4
        float4 v = *(const float4*)(xb + (size_t)c * NN + j4 * 4);
        float* dst = &lx[(c >> 1) * 128 + (c & 1)];
        dst[(j4*4+0)*2] = v.x; dst[(j4*4+1)*2] = v.y;
        dst[(j4*4+2)*2] = v.z; dst[(j4*4+3)*2] = v.w;
    }
    __syncthreads();

    const int w = tid >> 5, lane = tid & 31, half = lane >> 4, lj = lane & 15;
    const int m0 = 16 * w;
    const float* Wmat[3] = {gw, tw, pw};
    const float* Bv[3]   = {gbias, tbias, pbias};
    float*       Omat[3] = {G, T, P};

    #pragma unroll
    for (int p = 0; p < 3; ++p) {
        const float* A = Wmat[p] + (size_t)(m0 + lj) * CI + 2 * half;
        v8f a0 = {}, a1 = {}, a2 = {}, a3 = {};
        for (int k = 0; k < CI; k += 4) {
            v2f a = *(const v2f*)(A + k);                       // global b64
            const float* brow = &lx[((k >> 1) + half) * 128 + lj * 2];
            v2f b0 = *(const v2f*)(brow +  0);                  // ds b64
            v2f b1 = *(const v2f*)(brow + 32);
            v2f b2 = *(const v2f*)(brow + 64);
            v2f b3 = *(const v2f*)(brow + 96);
            a0 = wmma4(a, b0, a0);
            a1 = wmma4(a, b1, a1);
            a2 = wmma4(a, b2, a2);
            a3 = wmma4(a, b3, a3);
        }
        float* O = Omat[p] + (size_t)b * CO * NN + n0;
        #pragma unroll
        for (int r = 0; r < 8; ++r) {
            float bias = Bv[p][m0 + 8 * half + r];
            size_t row = (size_t)(m0 + 8 * half + r) * NN;
            O[row      + lj] = a0[r] + bias;
            O[row + 16 + lj] = a1[r] + bias;
            O[row + 32 + lj] = a2[r] + bias;
            O[row + 48 + lj] = a3[r] + bias;
        }
    }
}

// ---------------------------------------------------------------------------
// K2: split-K  Spart[b,ch] = G[b] * P[b]^T over m-chunk of 512.
// LDS-free: each lane streams one G/P row with b64 loads (lines fully
// consumed across k-iters; P rows shared by all 8 waves via WGP$).
// ---------------------------------------------------------------------------
__global__ __launch_bounds__(256) void s_partial_kernel(
    const float* __restrict__ G, const float* __restrict__ P,
    float* __restrict__ Spart)
{
    const int ch = blockIdx.x, b = blockIdx.y;
    const int tid = threadIdx.x, w = tid >> 5, lane = tid & 31;
    const int half = lane >> 4, lj = lane & 15;
    const float* Ga = G + ((size_t)b * CO + 16 * w + lj) * NN + 2 * half;
    const float* Pb = P + (size_t)b * CO * NN + 2 * half;

    v8f acc[8] = {};
    const int m0 = ch * 512;
    for (int m = m0; m < m0 + 512; m += 4) {
        v2f a = *(const v2f*)(Ga + m);
        #pragma unroll
        for (int ct = 0; ct < 8; ++ct) {
            v2f bf = *(const v2f*)(Pb + (size_t)(16 * ct + lj) * NN + m);
            acc[ct] = wmma4(a, bf, acc[ct]);
        }
    }
    float* out = Spart + (size_t)(b * 8 + ch) * CO * CO;
    #pragma unroll
    for (int ct = 0; ct < 8; ++ct)
        #pragma unroll
        for (int r = 0; r < 8; ++r)
            out[(size_t)(16 * w + 8 * half + r) * CO + 16 * ct + lj] = acc[ct][r];
}

// K2r: S = (sum over 8 chunks) / N
__global__ __launch_bounds__(256) void s_reduce_kernel(
    const float* __restrict__ Spart, float* __restrict__ S)
{
    int idx = blockIdx.x * 256 + threadIdx.x;        // 0..131071
    int b = idx >> 14, rem = idx & 16383;
    float s = 0.f;
    #pragma unroll
    for (int c = 0; c < 8; ++c)
        s += Spart[((size_t)(b * 8 + c) << 14) + rem];
    S[idx] = s * (1.0f / (float)NN);
}

// ---------------------------------------------------------------------------
// K3a: M2[b] = W_w @ S~[b]   (M=256,K=128,N=128) — tiny, LDS-free.
// ---------------------------------------------------------------------------
__global__ __launch_bounds__(256) void m2_kernel(
    const float* __restrict__ Ww, const float* __restrict__ S,
    float* __restrict__ M2)
{
    const int b = blockIdx.x;
    const int tid = threadIdx.x, w = tid >> 5, lane = tid & 31;
    const int half = lane >> 4, lj = lane & 15;
    const float* Sb = S + (size_t)b * CO * CO;
    float* M2b = M2 + (size_t)b * CI * CO;

    #pragma unroll
    for (int t = 0; t < 2; ++t) {
        const int m0 = 16 * (w + 8 * t);
        const float* A = Ww + (size_t)(m0 + lj) * CO + 2 * half;
        v8f acc[8] = {};
        for (int c = 0; c < CO; c += 4) {
            v2f a = *(const v2f*)(A + c);
            #pragma unroll
            for (int ct = 0; ct < 8; ++ct) {
                v2f bf;
                bf.x = Sb[(size_t)(c + 2 * half)     * CO + 16 * ct + lj];
                bf.y = Sb[(size_t)(c + 2 * half + 1) * CO + 16 * ct + lj];
                acc[ct] = wmma4(a, bf, acc[ct]);
            }
        }
        #pragma unroll
        for (int ct = 0; ct < 8; ++ct)
            #pragma unroll
            for (int r = 0; r < 8; ++r)
                M2b[(size_t)(m0 + 8 * half + r) * CO + 16 * ct + lj] = acc[ct][r];
    }
}

// ---------------------------------------------------------------------------
// K3b: U[b] = M2[b] @ T[b]  (M=256,K=128,N=4096) + per-block BN partials.
// T tile [128 x 64] pair-interleaved in LDS (stride 130 to stagger banks).
// ---------------------------------------------------------------------------
__global__ __launch_bounds__(256) void wy_kernel(
    const float* __restrict__ M2, const float* __restrict__ T,
    float* __restrict__ U, float* __restrict__ bnpart)
{
    extern __shared__ float lt[];                    // 64*130 floats
    const int b = blockIdx.y, n0 = blockIdx.x * 64, tid = threadIdx.x;
    const float* Tb = T + (size_t)b * CO * NN + n0;

    #pragma unroll
    for (int i = 0; i < 8; ++i) {                    // 2048 float4 loads
        int f4 = tid + 256 * i;
        int r  = f4 >> 4;                            // T row 0..127
        int j4 = f4 & 15;
        float4 v = *(const float4*)(Tb + (size_t)r * NN + j4 * 4);
        float* dst = &lt[(r >> 1) * 130 + (r & 1)];
        dst[(j4*4+0)*2] = v.x; dst[(j4*4+1)*2] = v.y;
        dst[(j4*4+2)*2] = v.z; dst[(j4*4+3)*2] = v.w;
    }
    __syncthreads();

    const int w = tid >> 5, lane = tid & 31, half = lane >> 4, lj = lane & 15;
    const int blk = blockIdx.y * 64 + blockIdx.x;    // 0..511
    const float* M2b = M2 + (size_t)b * CI * CO;
    float* Ub = U + (size_t)b * CI * NN + n0;

    #pragma unroll
    for (int t = 0; t < 2; ++t) {
        const int m0 = 16 * (w + 8 * t);
        const float* A = M2b + (size_t)(m0 + lj) * CO + 2 * half;
        v8f a0 = {}, a1 = {}, a2 = {}, a3 = {};
        for (int k = 0; k < CO; k += 4) {
            v2f a = *(const v2f*)(A + k);
            const float* brow = &lt[((k >> 1) + half) * 130 + lj * 2];
            v2f b0 = *(const v2f*)(brow +  0);
            v2f b1 = *(const v2f*)(brow + 32);
            v2f b2 = *(const v2f*)(brow + 64);
            v2f b3 = *(const v2f*)(brow + 96);
            a0 = wmma4(a, b0, a0);
            a1 = wmma4(a, b1, a1);
            a2 = wmma4(a, b2, a2);
            a3 = wmma4(a, b3, a3);
        }
        #pragma unroll
        for (int r = 0; r < 8; ++r) {
            size_t row = (size_t)(m0 + 8 * half + r) * NN;
            float v0 = a0[r], v1 = a1[r], v2 = a2[r], v3 = a3[r];
            Ub[row      + lj] = v0;
            Ub[row + 16 + lj] = v1;
            Ub[row + 32 + lj] = v2;
            Ub[row + 48 + lj] = v3;
            float ps = v0 + v1 + v2 + v3;
            float pq = v0*v0 + v1*v1 + v2*v2 + v3*v3;
            #pragma unroll
            for (int off = 8; off >= 1; off >>= 1) { // stays within 16-lane half
                ps += __shfl_xor(ps, off, 32);
                pq += __shfl_xor(pq, off, 32);
            }
            if (lj == 0) {
                int chan = m0 + 8 * half + r;        // unique per (wave,t,half,r)
                bnpart[((size_t)blk * CI + chan) * 2 + 0] = ps;
                bnpart[((size_t)blk * CI + chan) * 2 + 1] = pq;
            }
        }
    }
}

// K4: per-channel BN coefficients (double accumulation). W_b cancels.
__global__ __launch_bounds__(256) void bnstats_kernel(
    const float* __restrict__ bnpart,
    const float* __restrict__ gamma, const float* __restrict__ beta,
    float* __restrict__ coef)
{
    const int c = threadIdx.x;                       // 0..255
    double s = 0.0, q = 0.0;
    for (int blk = 0; blk < 512; ++blk) {
        s += (double)bnpart[((size_t)blk * CI + c) * 2 + 0];
        q += (double)bnpart[((size_t)blk * CI + c) * 2 + 1];
    }
    const double inv_n = 1.0 / (double)(BB * NN);
    double mean = s * inv_n;
    double var  = q * inv_n - mean * mean;
    double inv  = 1.0 / sqrt(var + 1e-5);
    double a    = (double)gamma[c] * inv;
    coef[2 * c + 0] = (float)a;
    coef[2 * c + 1] = (float)((double)beta[c] - mean * a);
}

// K5: out = U * a[c] + b[c] + x   (float4 streams)
__global__ __launch_bounds__(256) void final_kernel(
    const float* __restrict__ U, const float* __restrict__ x,
    const float* __restrict__ coef, float* __restrict__ out)
{
    const int t = blockIdx.x * 256 + threadIdx.x;    // 0..524287
    #pragma unroll
    for (int i = 0; i < 4; ++i) {
        size_t f4 = (size_t)t + (size_t)524288 * i;  // 0..2097151 vec4
        int c = (int)((f4 >> 10) & 255);             // 1024 vec4 per row
        float a  = coef[2 * c + 0];
        float b2 = coef[2 * c + 1];
        float4 u  = *(const float4*)(U + f4 * 4);
        float4 xv = *(const float4*)(x + f4 * 4);
        float4 o;
        o.x = fmaf(u.x, a, b2) + xv.x;
        o.y = fmaf(u.y, a, b2) + xv.y;
        o.z = fmaf(u.z, a, b2) + xv.z;
        o.w = fmaf(u.w, a, b2) + xv.w;
        *(float4*)(out + f4 * 4) = o;
    }
}

// ---------------------------------------------------------------------------
extern "C" void kernel_launch(void* const* d_in, const int* in_sizes, int n_in,
                              void* d_out, int out_size, void* d_ws, size_t ws_size,
                              hipStream_t stream)
{
    const float* x     = (const float*)d_in[0];
    const float* gw    = (const float*)d_in[1];
    const float* gb    = (const float*)d_in[2];
    const float* tw    = (const float*)d_in[3];
    const float* tb    = (const float*)d_in[4];
    const float* pw    = (const float*)d_in[5];
    const float* pb    = (const float*)d_in[6];
    const float* Ww    = (const float*)d_in[7];
    // d_in[8] = W_b : provably cancelled by training-mode BatchNorm
    const float* gamma = (const float*)d_in[9];
    const float* beta  = (const float*)d_in[10];

    float* ws    = (float*)d_ws;
    float* G     = ws + OFF_G;
    float* T     = ws + OFF_T;
    float* P     = ws + OFF_P;
    float* Spart = ws + OFF_SPART;
    float* S     = ws + OFF_S;
    float* M2    = ws + OFF_M2;
    float* U     = ws + OFF_U;
    float* bnp   = ws + OFF_BNP;
    float* coef  = ws + OFF_COEF;

    proj_kernel     <<<dim3(64, 8), 256, 65536, stream>>>(x, gw, gb, tw, tb, pw, pb, G, T, P);
    s_partial_kernel<<<dim3(8, 8),  256, 0,     stream>>>(G, P, Spart);
    s_reduce_kernel <<<512,         256, 0,     stream>>>(Spart, S);
    m2_kernel       <<<8,           256, 0,     stream>>>(Ww, S, M2);
    wy_kernel       <<<dim3(64, 8), 256, 33280, stream>>>(M2, T, U, bnp);
    bnstats_kernel  <<<1,           256, 0,     stream>>>(bnp, gamma, beta, coef);
    final_kernel    <<<2048,        256, 0,     stream>>>(U, x, coef, (float*)d_out);
}